// FeatureNet_24618752541388
// MI455X (gfx1250) — compile-verified
//
#include <hip/hip_runtime.h>
#include <cstdint>

typedef __attribute__((ext_vector_type(2))) float v2f;
typedef __attribute__((ext_vector_type(8))) float v8f;

#define TPB 256
#define GRID(n) dim3((unsigned)((((long long)(n)) + TPB - 1) / TPB))

static const int B_ = 8;
static const int N0_ = 8192;

// ---------------------------------------------------------------------------
// Farthest point sampling: one block per batch, dists live in LDS.
// ---------------------------------------------------------------------------
__global__ void fps_kernel(const float* __restrict__ xyz, int N, int npoint,
                           int* __restrict__ idx_out) {
    int b = blockIdx.x;
    const float* base = xyz + (size_t)b * N * 3;
    __shared__ float s_d[8192];
    __shared__ float red_v[TPB];
    __shared__ int   red_i[TPB];
    __shared__ int   s_far;
    int tid = threadIdx.x;
    for (int i = tid; i < N; i += TPB) s_d[i] = 1e10f;
    if (tid == 0) s_far = 0;
    __syncthreads();
    for (int s = 0; s < npoint; ++s) {
        int far = s_far;
        if (tid == 0) idx_out[b * npoint + s] = far;
        float cx = base[far*3+0], cy = base[far*3+1], cz = base[far*3+2];
        float bv = -1.0f; int bi = 0x7fffffff;
        for (int i = tid; i < N; i += TPB) {
            float dx = base[i*3+0]-cx, dy = base[i*3+1]-cy, dz = base[i*3+2]-cz;
            float d = dx*dx + dy*dy + dz*dz;
            d = fminf(s_d[i], d);
            s_d[i] = d;
            if (d > bv) { bv = d; bi = i; }
        }
        red_v[tid] = bv; red_i[tid] = bi;
        __syncthreads();
        for (int off = TPB/2; off > 0; off >>= 1) {
            if (tid < off) {
                float ov = red_v[tid+off]; int oi = red_i[tid+off];
                if (ov > red_v[tid] || (ov == red_v[tid] && oi < red_i[tid])) {
                    red_v[tid] = ov; red_i[tid] = oi;
                }
            }
            __syncthreads();
        }
        if (tid == 0) s_far = red_i[0];
        __syncthreads();
    }
}

__global__ void gather_xyz_kernel(const float* __restrict__ xyz, const int* __restrict__ fidx,
                                  float* __restrict__ nxyz, int N, int npoint, int B) {
    int t = blockIdx.x * blockDim.x + threadIdx.x;
    int total = B * npoint;
    if (t >= total) return;
    int b = t / npoint;
    int j = fidx[t];
    const float* p = xyz + ((size_t)b * N + j) * 3;
    nxyz[(size_t)t*3+0] = p[0];
    nxyz[(size_t)t*3+1] = p[1];
    nxyz[(size_t)t*3+2] = p[2];
}

// Ball query: sequential scan == "32 smallest in-radius indices", pad with first.
__global__ void ball_query_kernel(const float* __restrict__ xyz, const float* __restrict__ nxyz,
                                  int N, int npoint, int B, float r2, int ns,
                                  int* __restrict__ idx) {
    int t = blockIdx.x * blockDim.x + threadIdx.x;
    int total = B * npoint;
    if (t >= total) return;
    int b = t / npoint;
    const float* base = xyz + (size_t)b * N * 3;
    float cx = nxyz[(size_t)t*3+0], cy = nxyz[(size_t)t*3+1], cz = nxyz[(size_t)t*3+2];
    int* out = idx + (size_t)t * ns;
    int cnt = 0;
    for (int p = 0; p < N && cnt < ns; ++p) {
        float dx = base[p*3+0]-cx, dy = base[p*3+1]-cy, dz = base[p*3+2]-cz;
        float d = dx*dx + dy*dy + dz*dz;
        if (d < r2) out[cnt++] = p;
    }
    int fill = (cnt > 0) ? out[0] : 0;
    for (; cnt < ns; ++cnt) out[cnt] = fill;
}

// Build grouped input matrix G[R x Kpad]: [rel_xyz(3) | feats(Cf) | zero pad]
__global__ void group_kernel(const float* __restrict__ xyz, const float* __restrict__ feats,
                             const float* __restrict__ nxyz, const int* __restrict__ idx,
                             float* __restrict__ G, int N, int npoint, int B, int Cf, int Kpad) {
    long long t = (long long)blockIdx.x * blockDim.x + threadIdx.x;
    long long total = (long long)B * npoint * 32 * Kpad;
    if (t >= total) return;
    int c = (int)(t % Kpad);
    long long row = t / Kpad;
    long long bj = row >> 5;               // (b*npoint + j)
    int b = (int)(bj / npoint);
    int p = idx[row];
    float v;
    if (c < 3)            v = xyz[((size_t)b*N + p)*3 + c] - nxyz[bj*3 + c];
    else if (c < 3 + Cf)  v = feats[((size_t)b*N + p)*Cf + (c-3)];
    else                  v = 0.0f;
    G[(size_t)row * Kpad + c] = v;
}

// Zero-pad weights to Kpad rows.
__global__ void pad_w_kernel(const float* __restrict__ W, float* __restrict__ Wp,
                             int K, int Kpad, int C) {
    long long t = (long long)blockIdx.x * blockDim.x + threadIdx.x;
    long long total = (long long)Kpad * C;
    if (t >= total) return;
    int k = (int)(t / C), c = (int)(t % C);
    Wp[t] = (k < K) ? W[(size_t)k * C + c] : 0.0f;
}

// ---------------------------------------------------------------------------
// FP32 WMMA GEMM: Y[R x C] = X[R x K] * W[K x C]. One wave per 16x16 tile,
// K stepped by 4 via V_WMMA_F32_16X16X4_F32. R%16==0, C%16==0, K%4==0.
// ---------------------------------------------------------------------------
__global__ void gemm16_kernel(const float* __restrict__ X, const float* __restrict__ W,
                              float* __restrict__ Y, int R, int K, int C) {
    int gid  = blockIdx.x * blockDim.x + threadIdx.x;
    int wave = gid >> 5;
    int lane = threadIdx.x & 31;
    int tilesN = C >> 4;
    int tilesM = R >> 4;
    if (wave >= tilesM * tilesN) return;           // wave-uniform: EXEC stays full
    int tm = wave / tilesN, tn = wave % tilesN;
    int n  = lane & 15;                            // A row (M) and B col (N) index
    int kh = lane >> 4;                            // K half-select (0/1)
    const float* xp = X + ((size_t)(tm*16 + n)) * K + 2*kh;
    const float* wp = W + (size_t)(2*kh) * C + tn*16 + n;
    v8f acc = {};
    for (int k = 0; k < K; k += 4) {
        v2f a = *(const v2f*)(xp + k);             // A: M=lane%16, K = k + 2*kh + {0,1}
        v2f bf;
        bf.x = wp[(size_t)k * C];                  // B: N=lane%16, K = k + 2*kh
        bf.y = wp[(size_t)k * C + C];              //              K = k + 2*kh + 1
        acc = __builtin_amdgcn_wmma_f32_16x16x4_f32(
            false, a, false, bf, (short)0, acc, false, false);
    }
    union { v8f v; float f[8]; } u; u.v = acc;
    float* yp = Y + ((size_t)(tm*16 + 8*kh)) * C + tn*16 + n;   // D: M=v+8*kh, N=lane%16
#pragma unroll
    for (int v = 0; v < 8; ++v) yp[(size_t)v * C] = u.f[v];
}

// Per-channel sum / sum-of-squares: one block per channel (no atomics needed).
__global__ void colsum_kernel(const float* __restrict__ Y, int R, int C,
                              float* __restrict__ sum, float* __restrict__ sumsq) {
    int c = blockIdx.x;
    __shared__ float s1[TPB], s2[TPB];
    float a = 0.f, b = 0.f;
    for (int r = threadIdx.x; r < R; r += TPB) {
        float v = Y[(size_t)r * C + c];
        a += v; b += v * v;
    }
    s1[threadIdx.x] = a; s2[threadIdx.x] = b;
    __syncthreads();
    for (int off = TPB/2; off > 0; off >>= 1) {
        if (threadIdx.x < off) { s1[threadIdx.x] += s1[threadIdx.x+off];
                                 s2[threadIdx.x] += s2[threadIdx.x+off]; }
        __syncthreads();
    }
    if (threadIdx.x == 0) { sum[c] = s1[0]; sumsq[c] = s2[0]; }
}

__global__ void bn_finalize_kernel(const float* __restrict__ sum, const float* __restrict__ sumsq,
                                   const float* __restrict__ gamma, const float* __restrict__ beta,
                                   int C, float invR, float* __restrict__ scale,
                                   float* __restrict__ shift) {
    int c = blockIdx.x * blockDim.x + threadIdx.x;
    if (c >= C) return;
    float m  = sum[c] * invR;
    float vv = sumsq[c] * invR - m * m;
    float sc = gamma[c] * rsqrtf(vv + 1e-5f);
    scale[c] = sc;
    shift[c] = beta[c] - m * sc;
}

__global__ void bn_relu_kernel(const float* __restrict__ Yin, float* __restrict__ Yout,
                               long long total, int C, const float* __restrict__ scale,
                               const float* __restrict__ shift) {
    long long t = (long long)blockIdx.x * blockDim.x + threadIdx.x;
    if (t >= total) return;
    int c = (int)(t % C);
    float v = Yin[t] * scale[c] + shift[c];
    Yout[t] = v > 0.f ? v : 0.f;
}

__global__ void maxpool_kernel(const float* __restrict__ Y, float* __restrict__ out,
                               int BJ, int ns, int C) {
    long long t = (long long)blockIdx.x * blockDim.x + threadIdx.x;
    long long total = (long long)BJ * C;
    if (t >= total) return;
    long long bj = t / C; int c = (int)(t % C);
    const float* p = Y + ((size_t)bj * ns) * C + c;
    float m = p[0];
    for (int s = 1; s < ns; ++s) m = fmaxf(m, p[(size_t)s * C]);
    out[t] = m;
}

// 3-NN inverse-distance interpolation + concat into G[R x Kpad]: [interp(C2)|f1(C1)|pad]
__global__ void fp_interp_kernel(const float* __restrict__ xyz1, const float* __restrict__ xyz2,
                                 const float* __restrict__ f1, const float* __restrict__ f2,
                                 float* __restrict__ G, int N1, int N2, int B,
                                 int C1, int C2, int Kpad) {
    int t = blockIdx.x * blockDim.x + threadIdx.x;
    int total = B * N1;
    if (t >= total) return;
    int b = t / N1;
    const float* p1 = xyz1 + (size_t)t * 3;
    const float* x2 = xyz2 + (size_t)b * N2 * 3;
    float px = p1[0], py = p1[1], pz = p1[2];
    float d0 = 1e30f, d1 = 1e30f, d2 = 1e30f;
    int   i0 = 0, i1 = 0, i2 = 0;
    for (int j = 0; j < N2; ++j) {
        float dx = x2[j*3+0]-px, dy = x2[j*3+1]-py, dz = x2[j*3+2]-pz;
        float d = dx*dx + dy*dy + dz*dz;
        if (d < d0)      { d2=d1; i2=i1; d1=d0; i1=i0; d0=d; i0=j; }
        else if (d < d1) { d2=d1; i2=i1; d1=d;  i1=j; }
        else if (d < d2) { d2=d;  i2=j; }
    }
    float w0 = 1.f/(d0+1e-8f), w1 = 1.f/(d1+1e-8f), w2 = 1.f/(d2+1e-8f);
    float ws = w0 + w1 + w2; w0 /= ws; w1 /= ws; w2 /= ws;
    const float* f2b = f2 + (size_t)b * N2 * C2;
    float* g = G + (size_t)t * Kpad;
    for (int c = 0; c < C2; ++c)
        g[c] = w0*f2b[(size_t)i0*C2+c] + w1*f2b[(size_t)i1*C2+c] + w2*f2b[(size_t)i2*C2+c];
    const float* f1p = f1 + (size_t)t * C1;
    for (int c = 0; c < C1; ++c) g[C2 + c] = f1p[c];
    for (int c = C2 + C1; c < Kpad; ++c) g[c] = 0.f;
}

// ---------------------------------------------------------------------------
// Host side
// ---------------------------------------------------------------------------
struct LayerP { const float* W; const float* g; const float* b; int cin, cout; };

static bool map_all(void* const* d_in, const int* in_sizes, int n_in, bool strict, int variant,
                    const float** pts, LayerP* saL, LayerP* fpL, LayerP* fcL) {
    static const int sa_dims[4][4] = {{6,32,32,64},{67,64,64,128},{131,128,128,256},{259,256,256,512}};
    static const int sa_nl[4] = {3,3,3,3};
    static const int fp_dims[4][4] = {{131,128,128,128},{320,256,128,0},{384,256,256,0},{768,256,256,0}};
    static const int fp_nl[4] = {3,2,2,2};
    int pos = 0; bool ok = true;
    auto take = [&](int sz) -> const float* {
        if (pos >= n_in) { ok = false; return nullptr; }
        if (strict && in_sizes[pos] != sz) { ok = false; return nullptr; }
        return (const float*)d_in[pos++];
    };
    auto takeL = [&](LayerP* L, int cin, int cout, bool gfirst) {
        L->cin = cin; L->cout = cout;
        L->W = take(cin * cout);
        const float* x = take(cout);
        const float* y = take(cout);
        if (gfirst) { L->g = x; L->b = y; } else { L->b = x; L->g = y; }
    };
    auto doSA = [&](bool gf){ int li=0; for(int l=0;l<4;++l) for(int j=0;j<sa_nl[l];++j) takeL(&saL[li++], sa_dims[l][j], sa_dims[l][j+1], gf); };
    auto doFP = [&](bool gf){ int li=0; for(int l=0;l<4;++l) for(int j=0;j<fp_nl[l];++j) takeL(&fpL[li++], fp_dims[l][j], fp_dims[l][j+1], gf); };
    auto doFC = [&](bool gf){ takeL(&fcL[0], 128, 128, gf); };
    if (variant == 0) { *pts = take(B_*N0_*3); doSA(true);  doFP(true);  doFC(true); }
    else              { doFC(false); doFP(false); doSA(false); *pts = take(B_*N0_*3); }
    if (strict && pos != n_in) ok = false;
    return ok;
}

static const float* run_mlp(hipStream_t stream, const float* X0, int ldx0, int R,
                            const LayerP* Ls, int nl, float* bufA, float* bufB, float* Wp,
                            float* sumB, float* sqB, float* scB, float* shB, float* finalOut) {
    const float* X = X0; int ldx = ldx0;
    float* Y = bufB;
    for (int l = 0; l < nl; ++l) {
        int K = Ls[l].cin, C = Ls[l].cout;
        int Kp = (l == 0) ? ldx : K;                     // first-layer X is pre-padded
        pad_w_kernel<<<GRID((long long)Kp * C), TPB, 0, stream>>>(Ls[l].W, Wp, K, Kp, C);
        long long th = (long long)(R/16) * (C/16) * 32;
        gemm16_kernel<<<GRID(th), TPB, 0, stream>>>(X, Wp, Y, R, Kp, C);
        colsum_kernel<<<C, TPB, 0, stream>>>(Y, R, C, sumB, sqB);
        bn_finalize_kernel<<<GRID(C), TPB, 0, stream>>>(sumB, sqB, Ls[l].g, Ls[l].b, C,
                                                        1.0f/(float)R, scB, shB);
        long long tot = (long long)R * C;
        float* out = (l == nl-1 && finalOut) ? finalOut : Y;
        bn_relu_kernel<<<GRID(tot), TPB, 0, stream>>>(Y, out, tot, C, scB, shB);
        X = out; ldx = C;
        Y = (Y == bufB) ? bufA : bufB;
    }
    return X;
}

extern "C" void kernel_launch(void* const* d_in, const int* in_sizes, int n_in,
                              void* d_out, int out_size, void* d_ws, size_t ws_size,
                              hipStream_t stream) {
    (void)out_size; (void)ws_size;
    const float* points = nullptr;
    LayerP saL[12], fpL[9], fcL[1];
    if (!map_all(d_in, in_sizes, n_in, true, 0, &points, saL, fpL, fcL))
        if (!map_all(d_in, in_sizes, n_in, true, 1, &points, saL, fpL, fcL))
            map_all(d_in, in_sizes, n_in, false, 0, &points, saL, fpL, fcL);

    // Workspace bump allocator (floats, 16B aligned chunks)
    float* ws = (float*)d_ws;
    size_t off = 0;
    auto alloc = [&](size_t n) { float* p = ws + off; off += (n + 3) & ~(size_t)3; return p; };

    float* xyzL[5]; xyzL[0] = nullptr;
    xyzL[1] = alloc((size_t)B_*1024*3); xyzL[2] = alloc((size_t)B_*256*3);
    xyzL[3] = alloc((size_t)B_*64*3);   xyzL[4] = alloc((size_t)B_*16*3);
    float* saF[5]; saF[0] = nullptr;
    saF[1] = alloc((size_t)B_*1024*64);  saF[2] = alloc((size_t)B_*256*128);
    saF[3] = alloc((size_t)B_*64*256);   saF[4] = alloc((size_t)B_*16*512);
    float* fpF[4];
    fpF[3] = alloc((size_t)B_*64*256);   fpF[2] = alloc((size_t)B_*256*256);
    fpF[1] = alloc((size_t)B_*1024*128); fpF[0] = alloc((size_t)B_*8192*128);
    int* fpsIdx  = (int*)alloc((size_t)B_*1024);
    int* ballIdx = (int*)alloc((size_t)B_*1024*32);
    float* Wp   = alloc((size_t)768*512);
    float* sumB = alloc(512); float* sqB = alloc(512);
    float* scB  = alloc(512); float* shB = alloc(512);
    float* bufA = alloc((size_t)16777216);
    float* bufB = alloc((size_t)16777216);

    const int   levN[5] = {8192, 1024, 256, 64, 16};
    const int   levC[5] = {3, 64, 128, 256, 512};
    const int   sa_np[4] = {1024, 256, 64, 16};
    const float sa_r2[4] = {0.01f, 0.04f, 0.16f, 0.64f};

    const float* xyz_lv[5]  = {points, xyzL[1], xyzL[2], xyzL[3], xyzL[4]};
    const float* feat_lv[5] = {points, saF[1], saF[2], saF[3], saF[4]};

    // -------- Set abstraction levels --------
    int li = 0;
    for (int l = 0; l < 4; ++l) {
        int Nin = levN[l], np = sa_np[l];
        fps_kernel<<<B_, TPB, 0, stream>>>(xyz_lv[l], Nin, np, fpsIdx);
        gather_xyz_kernel<<<GRID(B_*np), TPB, 0, stream>>>(xyz_lv[l], fpsIdx, xyzL[l+1], Nin, np, B_);
        ball_query_kernel<<<GRID(B_*np), TPB, 0, stream>>>(xyz_lv[l], xyzL[l+1], Nin, np, B_,
                                                           sa_r2[l], 32, ballIdx);
        int Cf = levC[l];
        int Kp = (3 + Cf + 3) & ~3;
        long long R = (long long)B_ * np * 32;
        group_kernel<<<GRID(R * Kp), TPB, 0, stream>>>(xyz_lv[l], feat_lv[l], xyzL[l+1],
                                                       ballIdx, bufA, Nin, np, B_, Cf, Kp);
        const float* lastY = run_mlp(stream, bufA, Kp, (int)R, &saL[li], 3,
                                     bufA, bufB, Wp, sumB, sqB, scB, shB, nullptr);
        maxpool_kernel<<<GRID((long long)B_*np*levC[l+1]), TPB, 0, stream>>>(
            lastY, saF[l+1], B_*np, 32, levC[l+1]);
        li += 3;
    }

    // -------- Feature propagation (coarse -> fine) --------
    const int fpOff[4] = {0, 3, 5, 7};
    const int fpNL[4]  = {3, 2, 2, 2};
    for (int l = 3; l >= 0; --l) {
        int N1 = levN[l], N2 = levN[l+1];
        const float* f1 = (l == 0) ? points : saF[l];
        int C1 = levC[l];
        const float* f2; int C2;
        if (l == 3) { f2 = saF[4]; C2 = 512; }
        else        { f2 = fpF[l+1]; C2 = (l >= 1) ? 256 : 128; }
        int Kp = (C2 + C1 + 3) & ~3;
        fp_interp_kernel<<<GRID(B_*N1), TPB, 0, stream>>>(xyz_lv[l], xyz_lv[l+1], f1, f2,
                                                          bufA, N1, N2, B_, C1, C2, Kp);
        run_mlp(stream, bufA, Kp, B_*N1, &fpL[fpOff[l]], fpNL[l],
                bufA, bufB, Wp, sumB, sqB, scB, shB, fpF[l]);
    }

    // -------- Final FC MLP -> d_out --------
    run_mlp(stream, fpF[0], 128, B_*N0_, fcL, 1,
            bufA, bufB, Wp, sumB, sqB, scB, shB, (float*)d_out);
}